// AnalogBidirRNNLayer_38354057953354
// MI455X (gfx1250) — compile-verified
//
#include <hip/hip_runtime.h>

// Bidirectional LSTM, persistent-kernel recurrent scan, f32 WMMA (gfx1250).
// T=256, B=64, I=512, H=512.
//
// Grid: 256 WGs x 128 threads.  WG = (dir, b-tile, j-tile); wave w in WG owns
// gate w (i,f,g,o) 16x16 output tile.  Per step each wave runs 256
// v_wmma_f32_16x16x4_f32 over K = I+H = 1024, fed entirely from LDS, using
// 4 independent accumulators to pipeline the matrix unit.
//
// LDS (dynamic, exactly 320 KB):
//   [0 .. 256KB)      per-wave pre-packed Wi|Wh B-fragments (WMMA lane layout)
//   [256KB .. 320KB)  shared A-fragments, frag-packed: x part (kk 0..127),
//                     h part (kk 128..255).  4x16x16 gate staging (4KB)
//                     aliases this region after the K-loop consumes it.

#define T_  256
#define B_  64
#define I_  512
#define H_  512

typedef float v2f __attribute__((ext_vector_type(2)));
typedef float v8f __attribute__((ext_vector_type(8)));

#define WFRAG_DW   (4 * 256 * 32 * 2)   // 65536 dwords = 256 KB
#define AFRAG_DW   (256 * 32 * 2)       // 16384 dwords =  64 KB
#define LDS_BYTES  ((WFRAG_DW + AFRAG_DW) * 4)   // 327,680 B = 320 KB

__device__ __forceinline__ float fast_sigmoid(float v) {
  return 1.f / (1.f + __expf(-v));
}
__device__ __forceinline__ float fast_tanh(float v) {
  // branch-free: tanh(v) = 1 - 2/(e^{2v}+1); saturates correctly at +/-inf
  return 1.f - 2.f / (__expf(2.f * v) + 1.f);
}

__global__ __launch_bounds__(128) void bidir_lstm_persistent(
    const float* __restrict__ x,
    const float* __restrict__ h0_f, const float* __restrict__ c0_f,
    const float* __restrict__ h0_b, const float* __restrict__ c0_b,
    const float* __restrict__ Wi_f, const float* __restrict__ Wh_f,
    const float* __restrict__ bi_f, const float* __restrict__ bh_f,
    const float* __restrict__ Wi_b, const float* __restrict__ Wh_b,
    const float* __restrict__ bi_b, const float* __restrict__ bh_b,
    float* out,                 // [T,B,2H] ++ hT_f[B,H] ++ cT_f ++ hT_b ++ cT_b
    unsigned int* counters)     // [2] per-direction step-arrival counters (pre-zeroed)
{
  extern __shared__ float lds[];
  float* wfrag = lds;                 // packed weight (B) fragments, per wave
  float* afrag = lds + WFRAG_DW;      // packed activation (A) fragments, shared
  float* gates = afrag;               // alias: 4x16x16 gate staging (after K-loop)

  const int tid  = threadIdx.x;
  const int lane = tid & 31;
  const int wave = tid >> 5;          // gate index 0..3 (i,f,g,o)
  const int m    = lane & 15;         // A-row / B-col within tile
  const int hi   = lane >> 4;         // K sub-group (0 -> K 0,1 ; 1 -> K 2,3)

  const int wg  = blockIdx.x;         // 0..255
  const int dir = wg >> 7;            // 0 = forward, 1 = backward
  const int rem = wg & 127;
  const int bt  = rem >> 5;           // batch tile 0..3
  const int jt  = rem & 31;           // column tile within H, 0..31

  const float* Wi = dir ? Wi_b : Wi_f;
  const float* Wh = dir ? Wh_b : Wh_f;
  const float* bi = dir ? bi_b : bi_f;
  const float* bh = dir ? bh_b : bh_f;
  const float* h0 = dir ? h0_b : h0_f;
  const float* c0 = dir ? c0_b : c0_f;

  // ---- one-time: pack this wave's Wi|Wh B-fragments into LDS ----
  // B-frag for K-step kk, lane l: { W[n][4kk+2*hi], W[n][4kk+2*hi+1] }, n = l&15.
  {
    const int n_g = wave * H_ + jt * 16 + m;
    const float* wrow_i = Wi + (size_t)n_g * I_ + hi * 2;
    const float* wrow_h = Wh + (size_t)n_g * H_ + hi * 2;
    float* dst = wfrag + (size_t)wave * (256 * 32 * 2) + lane * 2;
    for (int kk = 0; kk < 128; ++kk)
      *(v2f*)(dst + kk * 64) = *(const v2f*)(wrow_i + kk * 4);
    for (int kk = 0; kk < 128; ++kk)
      *(v2f*)(dst + (kk + 128) * 64) = *(const v2f*)(wrow_h + kk * 4);
  }

  // ---- per-thread cell state: 2 elements of the 16x16 (b x j) tile ----
  int ee[2]  = { tid, tid + 128 };
  int rr_[2], cc_[2], bg_[2], jc_[2];
  float bsum[2][4], cst[2];
  #pragma unroll
  for (int q = 0; q < 2; ++q) {
    rr_[q] = ee[q] >> 4;  cc_[q] = ee[q] & 15;
    bg_[q] = bt * 16 + rr_[q];
    jc_[q] = jt * 16 + cc_[q];
    #pragma unroll
    for (int g = 0; g < 4; ++g)
      bsum[q][g] = bi[g * H_ + jc_[q]] + bh[g * H_ + jc_[q]];
    cst[q] = c0[bg_[q] * H_ + jc_[q]];
  }

  float* HF = out + (size_t)T_ * B_ * 2 * H_;
  float* CF = HF + B_ * H_;
  float* HB = CF + B_ * H_;
  float* CB = HB + B_ * H_;

  const float* wsrc = wfrag + (size_t)wave * (256 * 32 * 2) + lane * 2;
  const float* asrc = afrag + lane * 2;
  unsigned int* cnt = counters + dir;

  for (int s = 0; s < T_; ++s) {
    const int tx = dir ? (T_ - 1 - s) : s;

    __syncthreads();   // previous step's gate staging fully consumed

    // ---- stage x_t A-fragments (kk 0..127) into LDS, frag-packed ----
    // float4 (row mm, cols 4c..4c+3) -> frag kk=c, lanes (0,mm) and (1,mm):
    // two 8B stores 128B apart (ds_store_2addr_b64).
    {
      const float* xbase = x + ((size_t)tx * B_ + bt * 16) * I_;
      #pragma unroll
      for (int rep = 0; rep < 16; ++rep) {
        int idx = rep * 128 + tid;        // 2048 float4 total
        int c   = idx >> 4;               // frag index 0..127
        int mm  = idx & 15;               // row within tile
        float4 t4 = *(const float4*)(xbase + (size_t)mm * I_ + c * 4);
        float* d0 = afrag + (c * 32 + mm) * 2;
        *(v2f*)(d0)      = v2f{t4.x, t4.y};
        *(v2f*)(d0 + 32) = v2f{t4.z, t4.w};
      }
    }

    // ---- stage h_prev A-fragments (kk 128..255) into LDS, frag-packed ----
    {
      const float* hsrc;
      size_t hstride;
      if (s == 0) {
        hsrc = h0 + (size_t)bt * 16 * H_;
        hstride = H_;
      } else {
        const int tprev = dir ? (tx + 1) : (tx - 1);
        hsrc = out + ((size_t)tprev * B_ + bt * 16) * (2 * H_) + dir * H_;
        hstride = 2 * H_;
      }
      #pragma unroll
      for (int rep = 0; rep < 16; ++rep) {
        int idx = rep * 128 + tid;
        int c   = idx >> 4;
        int mm  = idx & 15;
        float4 t4 = *(const float4*)(hsrc + (size_t)mm * hstride + c * 4);
        float* d0 = afrag + ((c + 128) * 32 + mm) * 2;
        *(v2f*)(d0)      = v2f{t4.x, t4.y};
        *(v2f*)(d0 + 32) = v2f{t4.z, t4.w};
      }
    }
    __syncthreads();

    // hint: prefetch next step's x rows while we compute this step
    if (s + 1 < T_) {
      const int txn = dir ? (tx - 1) : (tx + 1);
      __builtin_prefetch(x + ((size_t)txn * B_ + bt * 16 + m) * I_ + hi * 256, 0, 1);
    }

    // ---- gates tile = x_t @ Wi^T + h_prev @ Wh^T ----
    // 256 f32 WMMAs from LDS; 4 independent accumulators break the C->C
    // dependency chain so the matrix pipe runs at throughput, not latency.
    v8f a0 = {0.f,0.f,0.f,0.f,0.f,0.f,0.f,0.f};
    v8f a1 = a0, a2 = a0, a3 = a0;
    #pragma unroll 2
    for (int kk = 0; kk < 256; kk += 4) {
      v2f aa0 = *(const v2f*)(asrc + (kk + 0) * 64);
      v2f bb0 = *(const v2f*)(wsrc + (kk + 0) * 64);
      v2f aa1 = *(const v2f*)(asrc + (kk + 1) * 64);
      v2f bb1 = *(const v2f*)(wsrc + (kk + 1) * 64);
      v2f aa2 = *(const v2f*)(asrc + (kk + 2) * 64);
      v2f bb2 = *(const v2f*)(wsrc + (kk + 2) * 64);
      v2f aa3 = *(const v2f*)(asrc + (kk + 3) * 64);
      v2f bb3 = *(const v2f*)(wsrc + (kk + 3) * 64);
      a0 = __builtin_amdgcn_wmma_f32_16x16x4_f32(false, aa0, false, bb0,
                                                 (short)0, a0, false, false);
      a1 = __builtin_amdgcn_wmma_f32_16x16x4_f32(false, aa1, false, bb1,
                                                 (short)0, a1, false, false);
      a2 = __builtin_amdgcn_wmma_f32_16x16x4_f32(false, aa2, false, bb2,
                                                 (short)0, a2, false, false);
      a3 = __builtin_amdgcn_wmma_f32_16x16x4_f32(false, aa3, false, bb3,
                                                 (short)0, a3, false, false);
    }
    v8f acc = (a0 + a1) + (a2 + a3);

    __syncthreads();   // all waves done reading afrag -> safe to alias gates
    // ---- stage gate tiles: gates[gate*256 + row*16 + col] ----
    #pragma unroll
    for (int v = 0; v < 8; ++v) {
      int row = v + hi * 8;              // C layout: VGPR v -> M = v + 8*(lane>=16)
      gates[wave * 256 + row * 16 + m] = acc[v];
    }
    __syncthreads();

    // ---- elementwise LSTM cell (2 elements / thread), branch-free ----
    #pragma unroll
    for (int q = 0; q < 2; ++q) {
      int base = rr_[q] * 16 + cc_[q];
      float gi = gates[0 * 256 + base] + bsum[q][0];
      float gf = gates[1 * 256 + base] + bsum[q][1];
      float gg = gates[2 * 256 + base] + bsum[q][2];
      float go = gates[3 * 256 + base] + bsum[q][3];
      float i_ = fast_sigmoid(gi);
      float f_ = fast_sigmoid(gf);
      float g_ = fast_tanh(gg);
      float o_ = fast_sigmoid(go);
      float c_ = f_ * cst[q] + i_ * g_;
      cst[q] = c_;
      float h_ = o_ * fast_tanh(c_);
      out[((size_t)tx * B_ + bg_[q]) * (2 * H_) + dir * H_ + jc_[q]] = h_;
      if (s == T_ - 1) {
        (dir ? HB : HF)[bg_[q] * H_ + jc_[q]] = h_;
        (dir ? CB : CF)[bg_[q] * H_ + jc_[q]] = cst[q];
      }
    }

    // ---- device-wide per-direction step barrier ----
    __threadfence();
    __syncthreads();
    if (tid == 0) {
      __hip_atomic_fetch_add(cnt, 1u, __ATOMIC_RELEASE, __HIP_MEMORY_SCOPE_AGENT);
      const unsigned int target = (unsigned int)(s + 1) * 128u;
      while (__hip_atomic_load(cnt, __ATOMIC_ACQUIRE, __HIP_MEMORY_SCOPE_AGENT) < target) {
        __builtin_amdgcn_s_sleep(2);
      }
    }
    __syncthreads();
  }
}

extern "C" void kernel_launch(void* const* d_in, const int* in_sizes, int n_in,
                              void* d_out, int out_size, void* d_ws, size_t ws_size,
                              hipStream_t stream) {
  const float* x    = (const float*)d_in[0];
  const float* h0_f = (const float*)d_in[1];
  const float* c0_f = (const float*)d_in[2];
  const float* h0_b = (const float*)d_in[3];
  const float* c0_b = (const float*)d_in[4];
  const float* Wi_f = (const float*)d_in[5];
  const float* Wh_f = (const float*)d_in[6];
  const float* bi_f = (const float*)d_in[7];
  const float* bh_f = (const float*)d_in[8];
  const float* Wi_b = (const float*)d_in[9];
  const float* Wh_b = (const float*)d_in[10];
  const float* bi_b = (const float*)d_in[11];
  const float* bh_b = (const float*)d_in[12];

  unsigned int* counters = (unsigned int*)d_ws;
  hipMemsetAsync(d_ws, 0, 64, stream);   // zero the 2 per-direction counters

  dim3 grid(256), block(128);
  bidir_lstm_persistent<<<grid, block, LDS_BYTES, stream>>>(
      x, h0_f, c0_f, h0_b, c0_b,
      Wi_f, Wh_f, bi_f, bh_f,
      Wi_b, Wh_b, bi_b, bh_b,
      (float*)d_out, counters);
}